// SparseActivation_85864986182245
// MI455X (gfx1250) — compile-verified
//
#include <hip/hip_runtime.h>
#include <stdint.h>

// Row-wise top-k (k = D/2) masking: out = x * mask(top-k per row).
// One workgroup per 4096-float row. Async global->LDS streaming (CDNA5 TDM-class
// async ops, ASYNCcnt), in-LDS 8-bit MSB radix select, async LDS->global store.

#define D      4096
#define KTOP   2048
#define NT     256
#define PER    (D / NT)      // 16 elements per thread
#define NWAVE  (NT / 32)     // 8 wave32s per block

__device__ __forceinline__ unsigned f2key(float f) {
    // Monotonic mapping: larger float (incl. sign) -> larger unsigned key.
    unsigned u = __float_as_uint(f);
    return (u & 0x80000000u) ? ~u : (u | 0x80000000u);
}

// In-place inclusive Hillis-Steele scan over s[0..255]; all 256 threads call.
__device__ __forceinline__ void scan256(unsigned* s, int tid) {
    #pragma unroll
    for (int step = 1; step < NT; step <<= 1) {
        unsigned a = s[tid];
        unsigned b = (tid >= step) ? s[tid - step] : 0u;
        __syncthreads();
        s[tid] = a + b;
        __syncthreads();
    }
}

__global__ __launch_bounds__(NT) void topk_half_mask_kernel(
        const float* __restrict__ x, float* __restrict__ out) {
    __shared__ __align__(16) float s_row[D];         // 16 KB row buffer
    __shared__ unsigned s_hist[NWAVE * 256];         // per-wave histograms, 8 KB
    __shared__ unsigned s_scan[NT];                  // scan scratch
    __shared__ unsigned s_sel[3];                    // [0]=prefix, [1]=kk, [2]=count_greater

    const int tid = threadIdx.x;
    const int row = blockIdx.x;
    // addrspace(3)->generic cast keeps the wave-relative LDS byte offset in the
    // low 32 bits (aperture lives entirely in addr[63:32] on CDNA5).
    const unsigned lds0 = (unsigned)(unsigned long long)(void*)&s_row[0];

    // ---- async global -> LDS load of the row (B128 per lane, coalesced) ----
    {
        unsigned long long gbase = (unsigned long long)(x + (size_t)row * D);
        #pragma unroll
        for (int j = 0; j < PER / 4; ++j) {
            unsigned off = (unsigned)((j * NT + tid) * 16);
            asm volatile("global_load_async_to_lds_b128 %0, %1, %2"
                         :: "v"(lds0 + off), "v"(off), "s"(gbase)
                         : "memory");
        }
        asm volatile("s_wait_asynccnt 0" ::: "memory");
    }
    __syncthreads();

    // ---- MSB radix select: find key T = k-th largest key in the row ----
    unsigned prefix = 0;
    unsigned kk = KTOP;                 // remaining rank from the top
    #pragma unroll
    for (int pass = 0; pass < 4; ++pass) {
        const int shift = 24 - pass * 8;
        const unsigned himask = (pass == 0) ? 0u : ~((1u << (32 - pass * 8)) - 1u);

        #pragma unroll
        for (int j = 0; j < NWAVE; ++j) s_hist[j * NT + tid] = 0u;
        __syncthreads();

        unsigned* myhist = &s_hist[(tid >> 5) * 256];   // per-wave copy (wave32)
        #pragma unroll
        for (int j = 0; j < PER; ++j) {
            unsigned key = f2key(s_row[j * NT + tid]);  // conflict-free LDS reads
            if ((key & himask) == prefix)
                atomicAdd(&myhist[(key >> shift) & 0xFFu], 1u);
        }
        __syncthreads();

        unsigned h = 0;                                 // total count for bin `tid`
        #pragma unroll
        for (int w = 0; w < NWAVE; ++w) h += s_hist[w * 256 + tid];

        s_scan[255 - tid] = h;                          // reversed -> suffix sums
        __syncthreads();
        scan256(s_scan, tid);
        unsigned incl = s_scan[255 - tid];              // sum of bins >= tid
        unsigned excl = incl - h;                       // sum of bins >  tid
        if (excl < kk && kk <= incl) {                  // unique winning digit
            s_sel[0] = prefix | ((unsigned)tid << shift);
            s_sel[1] = kk - excl;
        }
        __syncthreads();
        prefix = s_sel[0];
        kk = s_sel[1];
    }
    const unsigned T = prefix;                          // key of k-th largest

    // ---- counts for exact tie-break (lowest index first, like lax.top_k) ----
    if (tid == 0) s_sel[2] = 0u;
    __syncthreads();

    const int base = tid * PER;                         // contiguous chunk (index order)
    unsigned lg = 0, le = 0;
    #pragma unroll
    for (int j = 0; j < PER; ++j) {
        unsigned key = f2key(s_row[base + j]);
        lg += (key > T) ? 1u : 0u;
        le += (key == T) ? 1u : 0u;
    }
    atomicAdd(&s_sel[2], lg);
    s_scan[tid] = le;
    __syncthreads();
    scan256(s_scan, tid);                               // inclusive scan of equal-counts
    unsigned rank  = s_scan[tid] - le;                  // equals before my chunk
    const unsigned quota = KTOP - s_sel[2];             // equals to keep, by index

    // ---- apply mask in place in LDS ----
    #pragma unroll
    for (int j = 0; j < PER; ++j) {
        float v = s_row[base + j];
        unsigned key = f2key(v);
        bool keep = (key > T) || ((key == T) && (rank < quota));
        rank += (key == T) ? 1u : 0u;
        if (!keep) s_row[base + j] = 0.0f;
    }
    __syncthreads();

    // ---- async LDS -> global store of the masked row ----
    {
        unsigned long long gbase = (unsigned long long)(out + (size_t)row * D);
        #pragma unroll
        for (int j = 0; j < PER / 4; ++j) {
            unsigned off = (unsigned)((j * NT + tid) * 16);
            asm volatile("global_store_async_from_lds_b128 %0, %1, %2"
                         :: "v"(off), "v"(lds0 + off), "s"(gbase)
                         : "memory");
        }
        asm volatile("s_wait_asynccnt 0" ::: "memory");
    }
}

extern "C" void kernel_launch(void* const* d_in, const int* in_sizes, int n_in,
                              void* d_out, int out_size, void* d_ws, size_t ws_size,
                              hipStream_t stream) {
    const float* x = (const float*)d_in[0];
    float* out = (float*)d_out;
    const int rows = in_sizes[0] / D;                   // (4*4096*4096)/4096 = 16384
    topk_half_mask_kernel<<<dim3(rows), dim3(NT), 0, stream>>>(x, out);
}